// BaselineGCN_78597901517272
// MI455X (gfx1250) — compile-verified
//
#include <hip/hip_runtime.h>

// ---------------------------------------------------------------------------
// GCN forward for MI455X (gfx1250, wave32).
// Dense GEMM (h @ W2 + b2) uses V_WMMA_F32_16X16X4_F32 (full fp32 precision).
// SpMM via edge-parallel gather + global_atomic_add_f32 (L2-resident, 192MB).
// ---------------------------------------------------------------------------

typedef __attribute__((ext_vector_type(2))) float v2f;
typedef __attribute__((ext_vector_type(8))) float v8f;

#define HID 64
#define BN_EPS 1e-5f

// ---------------------------------------------------------------- fill ------
__global__ void gcn_fill_kernel(float* __restrict__ p, long long n, float val) {
    long long i = (long long)blockIdx.x * blockDim.x + threadIdx.x;
    if (i < n) p[i] = val;
}

// ------------------------------------------------------- h0 = x@W1 + b1 -----
__global__ void gcn_mlp1_kernel(const float* __restrict__ x,
                                const float* __restrict__ W1,   // [3,64]
                                const float* __restrict__ b1,
                                float* __restrict__ out,        // [n,64]
                                int n) {
    long long idx = (long long)blockIdx.x * blockDim.x + threadIdx.x;
    if (idx >= (long long)n * HID) return;
    int node = (int)(idx >> 6);
    int c    = (int)(idx & 63);
    const float* xr = x + (size_t)node * 3;
    float acc = b1[c];
    acc = fmaf(xr[0], W1[0 * HID + c], acc);
    acc = fmaf(xr[1], W1[1 * HID + c], acc);
    acc = fmaf(xr[2], W1[2 * HID + c], acc);
    out[idx] = acc;
}

// --------------------------------------- out[row] += vals * h[col]  (SpMM) --
// 16 lanes per edge; each lane handles 4 contiguous features (float4 gather).
__global__ void gcn_spmm_kernel(const int* __restrict__ row,
                                const int* __restrict__ col,
                                const float* __restrict__ vals,
                                const float* __restrict__ h,    // [n,64]
                                float* __restrict__ out,        // [n,64] pre-zeroed
                                int nEdges) {
    long long t = (long long)blockIdx.x * blockDim.x + threadIdx.x;
    int e = (int)(t >> 4);
    if (e >= nEdges) return;
    int q = ((int)t & 15) * 4;
    int r = row[e];
    int c = col[e];
    float v = vals[e];
    const float4 hv = *(const float4*)(h + (size_t)c * HID + q);
    float* dst = out + (size_t)r * HID + q;
    atomicAdd(dst + 0, v * hv.x);
    atomicAdd(dst + 1, v * hv.y);
    atomicAdd(dst + 2, v * hv.z);
    atomicAdd(dst + 3, v * hv.w);
}

// ---------------------------------------------- in-place BN(eval) + ReLU ----
__global__ void gcn_bnrelu_kernel(float* __restrict__ h,
                                  const float* __restrict__ g,
                                  const float* __restrict__ be,
                                  const float* __restrict__ m,
                                  const float* __restrict__ v,
                                  long long total) {
    long long idx = (long long)blockIdx.x * blockDim.x + threadIdx.x;
    if (idx >= total) return;
    int c = (int)(idx & 63);
    float s  = g[c] * rsqrtf(v[c] + BN_EPS);
    float sh = be[c] - m[c] * s;
    float val = fmaf(h[idx], s, sh);
    h[idx] = val > 0.0f ? val : 0.0f;
}

// ----------------------- pack W2 into WMMA-B friendly (K-pair, N) float2 ----
// w2p[2*(kk*64+n)] = W2[2kk][n], w2p[2*(kk*64+n)+1] = W2[2kk+1][n]
__global__ void gcn_packw2_kernel(const float* __restrict__ W2,  // [64,64]
                                  float* __restrict__ w2p) {
    int idx = blockIdx.x * blockDim.x + threadIdx.x;   // 0..2047
    if (idx >= 32 * HID) return;
    int kk = idx >> 6;
    int n  = idx & 63;
    w2p[2 * idx + 0] = W2[(2 * kk + 0) * HID + n];
    w2p[2 * idx + 1] = W2[(2 * kk + 1) * HID + n];
}

// -------------------- out = h @ W2 + b2 via V_WMMA_F32_16X16X4_F32 ----------
// One wave computes a 16-node x 64-col block: 4 accumulator tiles, 16 K-steps.
// A-frag (16x4 f32): lane L (m = L&15): k-base = k0 + (L>=16 ? 2 : 0),
//   contiguous b64 load from h[m][kbase..kbase+1].
// B-frag (4x16 f32): lane L (n = t*16 + (L&15)): rows K=k0+(L>=16?2:0),+1
//   -> one b64 load from the packed w2p.
// C/D (16x16 f32): VGPR i holds M = i + (L>=16 ? 8 : 0), N = L&15.
__global__ void gcn_gemm_wmma_kernel(const float* __restrict__ h,    // [n,64]
                                     const float* __restrict__ w2p,  // packed 64x64
                                     const float* __restrict__ b2,
                                     float* __restrict__ out,        // [n,64]
                                     int n) {
    int gtid = blockIdx.x * blockDim.x + threadIdx.x;
    int wave = gtid >> 5;
    int lane = threadIdx.x & 31;
    int half = lane >> 4;        // 0: lanes 0-15, 1: lanes 16-31
    int l16  = lane & 15;

    int mBase = wave * 16;
    int mRow  = mBase + l16;
    int mClamped = mRow < n ? mRow : (n - 1);        // guard OOB loads
    const float* aRow = h + (size_t)mClamped * HID;

    v8f acc[4];
#pragma unroll
    for (int t = 0; t < 4; ++t) {
        float bv = b2[t * 16 + l16];                 // bias depends on N only
#pragma unroll
        for (int i = 0; i < 8; ++i) acc[t][i] = bv;
    }

#pragma unroll
    for (int k0 = 0; k0 < HID; k0 += 4) {
        v2f a = *(const v2f*)(aRow + k0 + half * 2);
        int kk = (k0 >> 1) + half;                   // packed K-pair row
#pragma unroll
        for (int t = 0; t < 4; ++t) {
            v2f b = *(const v2f*)(w2p + 2 * (kk * HID + t * 16 + l16));
            acc[t] = __builtin_amdgcn_wmma_f32_16x16x4_f32(
                false, a, false, b, (short)0, acc[t], false, false);
        }
    }

#pragma unroll
    for (int t = 0; t < 4; ++t) {
        int colIdx = t * 16 + l16;
#pragma unroll
        for (int i = 0; i < 8; ++i) {
            int m = mBase + i + half * 8;
            if (m < n) out[(size_t)m * HID + colIdx] = acc[t][i];
        }
    }
}

// ------------- BN2 + ReLU + column-wise sum & max pooling (fused) -----------
// Post-ReLU values are >= 0, so uint-bit compare == float compare; max buffer
// initialized to 0x00000000 (+0.0f) is exactly relu's lower bound.
__global__ void gcn_pool_kernel(const float* __restrict__ h,   // [n,64]
                                const float* __restrict__ g,
                                const float* __restrict__ be,
                                const float* __restrict__ m,
                                const float* __restrict__ v,
                                float* __restrict__ colSum,    // [64]
                                unsigned int* __restrict__ colMax, // [64] bits
                                int n) {
    int tid = blockIdx.x * blockDim.x + threadIdx.x;
    int c = tid & 63;
    int nodeStart = tid >> 6;
    int stride = (gridDim.x * blockDim.x) >> 6;
    float s  = g[c] * rsqrtf(v[c] + BN_EPS);
    float sh = be[c] - m[c] * s;
    float acc = 0.0f, amax = 0.0f;
    for (int node = nodeStart; node < n; node += stride) {
        float val = fmaf(h[(size_t)node * HID + c], s, sh);
        val = val > 0.0f ? val : 0.0f;
        acc += val;
        amax = val > amax ? val : amax;
    }
    atomicAdd(colSum + c, acc);
    atomicMax(colMax + c, __float_as_uint(amax));
}

// ------------------- out[3] = concat(mean, max) @ Wc + bc -------------------
__global__ void gcn_final_kernel(const float* __restrict__ colSum,
                                 const unsigned int* __restrict__ colMax,
                                 const float* __restrict__ Wc,   // [128,3]
                                 const float* __restrict__ bc,
                                 float* __restrict__ out, int n) {
    __shared__ float part[3 * 128];
    int j = threadIdx.x;                                      // 0..127
    float gj = (j < HID) ? (colSum[j] / (float)n)
                         : __uint_as_float(colMax[j - HID]);
    for (int c = 0; c < 3; ++c) part[c * 128 + j] = gj * Wc[j * 3 + c];
    __syncthreads();
    if (j < 3) {
        float a = bc[j];
        for (int k = 0; k < 128; ++k) a += part[j * 128 + k];
        out[j] = a;
    }
}

// ---------------------------------------------------------------------------
extern "C" void kernel_launch(void* const* d_in, const int* in_sizes, int n_in,
                              void* d_out, int out_size, void* d_ws, size_t ws_size,
                              hipStream_t stream) {
    const float* x    = (const float*)d_in[0];
    const int*   row  = (const int*)  d_in[1];
    const int*   col  = (const int*)  d_in[2];
    const float* vals = (const float*)d_in[3];
    const float* W1   = (const float*)d_in[4];
    const float* b1   = (const float*)d_in[5];
    const float* g1   = (const float*)d_in[6];
    const float* be1  = (const float*)d_in[7];
    const float* m1   = (const float*)d_in[8];
    const float* v1   = (const float*)d_in[9];
    const float* W2   = (const float*)d_in[10];
    const float* b2   = (const float*)d_in[11];
    const float* g2   = (const float*)d_in[12];
    const float* be2  = (const float*)d_in[13];
    const float* m2   = (const float*)d_in[14];
    const float* v2   = (const float*)d_in[15];
    const float* Wc   = (const float*)d_in[16];
    const float* bc   = (const float*)d_in[17];
    float* out = (float*)d_out;

    const int n = in_sizes[0] / 3;   // 100,000 nodes
    const int e = in_sizes[1];       // 1,600,000 edges

    // workspace layout
    float* A    = (float*)d_ws;                   // [n,64] activations ping
    float* B    = A + (size_t)n * HID;            // [n,64] activations pong
    float* w2p  = B + (size_t)n * HID;            // 64*64 packed W2
    float* psum = w2p + HID * HID;                // 64 col sums
    unsigned int* pmax = (unsigned int*)(psum + HID); // 64 col maxes (bits)

    const long long feat = (long long)n * HID;
    const int featBlocks = (int)((feat + 255) / 256);
    const long long et   = (long long)e * 16;
    const int edgeBlocks = (int)((et + 255) / 256);
    const int waves      = (n + 15) / 16;
    const int gemmBlocks = (waves + 7) / 8;       // 8 waves (256 thr) per block

    // layer 1: h0 = x@W1+b1 ; h1 = SpMM(h0)
    gcn_mlp1_kernel<<<featBlocks, 256, 0, stream>>>(x, W1, b1, A, n);
    gcn_fill_kernel<<<featBlocks, 256, 0, stream>>>(B, feat, 0.0f);
    gcn_spmm_kernel<<<edgeBlocks, 256, 0, stream>>>(row, col, vals, A, B, e);
    gcn_bnrelu_kernel<<<featBlocks, 256, 0, stream>>>(B, g1, be1, m1, v1, feat);

    // layer 2: h2 = SpMM(h1@W2+b2)   (GEMM on the WMMA pipe)
    gcn_packw2_kernel<<<(32 * HID + 255) / 256, 256, 0, stream>>>(W2, w2p);
    gcn_gemm_wmma_kernel<<<gemmBlocks, 256, 0, stream>>>(B, w2p, b2, A, n);
    gcn_fill_kernel<<<featBlocks, 256, 0, stream>>>(B, feat, 0.0f);
    gcn_spmm_kernel<<<edgeBlocks, 256, 0, stream>>>(row, col, vals, A, B, e);

    // BN2 + ReLU + pooling, then classifier
    gcn_fill_kernel<<<1, 256, 0, stream>>>(psum, 2 * HID, 0.0f); // sums + max bits
    gcn_pool_kernel<<<256, 256, 0, stream>>>(B, g2, be2, m2, v2, psum, pmax, n);
    gcn_final_kernel<<<1, 128, 0, stream>>>(psum, pmax, Wc, bc, out, n);
}